// FilterbankEdge_26645977104527
// MI455X (gfx1250) — compile-verified
//
#include <hip/hip_runtime.h>

typedef __attribute__((ext_vector_type(2))) float v2f;
typedef __attribute__((ext_vector_type(8))) float v8f;

#define QDIM 21
#define NDIM 1368
#define OUT_COLS 64000   // 338*64 + 167*128 + 82*256

// One wave (32 threads) per block. Each block owns one window position p and a
// 16-wide channel tile; it computes all 64 batch rows as four 16x16 WMMA tiles,
// so every weight element is fetched exactly once per block (-> once from HBM).
template<int W, int S, int D, int P, int LOGW>
__global__ __launch_bounds__(32)
void fb_wmma_kernel(const float* __restrict__ x,
                    const float* __restrict__ wt,
                    float* __restrict__ out, int out_off) {
  constexpr int F = QDIM * W;          // K dimension of the per-position GEMM
  const int p    = blockIdx.x;
  const int nt   = blockIdx.y;         // channel tile (16 channels)
  const int start = (p == P - 1) ? (NDIM - W) : p * S;   // reference edge rule

  const int lane = threadIdx.x & 31;
  const int half = lane >> 4;          // selects K-pair (K0,K1) vs (K2,K3)
  const int lr   = lane & 15;          // M (for A) / N (for B) within tile

  // B-matrix fragment source: W.reshape(P, D, F)[p, nt*16+lr, k0 + 2*half + {0,1}]
  const float* __restrict__ bptr =
      wt + (size_t)p * D * F + (size_t)(nt * 16 + lr) * F + 2 * half;

  // A-matrix fragment source: x[(mt*16+lr), f/W, start + f%W], f = k0 + 2*half
  const size_t mstride = (size_t)16 * QDIM * NDIM;
  const float* __restrict__ x0 = x + (size_t)lr * (QDIM * NDIM) + start;

  v8f c0 = {}, c1 = {}, c2 = {}, c3 = {};

#pragma unroll 2
  for (int k0 = 0; k0 < F; k0 += 4) {
    const int f0  = k0 + 2 * half;
    // w is a power of two: q = f0 >> LOGW, j = f0 & (W-1); f0 even -> the K-pair
    // (f0, f0+1) stays inside one q-row and one 8B-aligned float2.
    const int off = ((f0 >> LOGW) * NDIM) + (f0 & (W - 1));

    const v2f bf = *(const v2f*)(bptr + k0);
    const v2f a0 = *(const v2f*)(x0 + off);
    const v2f a1 = *(const v2f*)(x0 + mstride + off);
    const v2f a2 = *(const v2f*)(x0 + 2 * mstride + off);
    const v2f a3 = *(const v2f*)(x0 + 3 * mstride + off);

    // D = A*B + C, full f32: v_wmma_f32_16x16x4_f32
    c0 = __builtin_amdgcn_wmma_f32_16x16x4_f32(false, a0, false, bf, (short)0, c0, false, false);
    c1 = __builtin_amdgcn_wmma_f32_16x16x4_f32(false, a1, false, bf, (short)0, c1, false, false);
    c2 = __builtin_amdgcn_wmma_f32_16x16x4_f32(false, a2, false, bf, (short)0, c2, false, false);
    c3 = __builtin_amdgcn_wmma_f32_16x16x4_f32(false, a3, false, bf, (short)0, c3, false, false);
  }

  // C/D layout: VGPR r holds batch (r + 8*half), channel = lane%16.
  float* __restrict__ obase = out + out_off + (size_t)p * D + nt * 16 + lr;
#pragma unroll
  for (int r = 0; r < 8; ++r) {
    const int b0 = r + 8 * half;
    obase[(size_t)(b0)      * OUT_COLS] = c0[r];
    obase[(size_t)(b0 + 16) * OUT_COLS] = c1[r];
    obase[(size_t)(b0 + 32) * OUT_COLS] = c2[r];
    obase[(size_t)(b0 + 48) * OUT_COLS] = c3[r];
  }
}

extern "C" void kernel_launch(void* const* d_in, const int* in_sizes, int n_in,
                              void* d_out, int out_size, void* d_ws, size_t ws_size,
                              hipStream_t stream) {
  (void)in_sizes; (void)n_in; (void)out_size; (void)d_ws; (void)ws_size;
  const float* x  = (const float*)d_in[0];
  const float* w0 = (const float*)d_in[1];
  const float* w1 = (const float*)d_in[2];
  const float* w2 = (const float*)d_in[3];
  float* out = (float*)d_out;

  // fb0: w=16 s=4  d=64  P=338  -> cols [0, 21632)
  fb_wmma_kernel<16, 4, 64, 338, 4><<<dim3(338, 4),  32, 0, stream>>>(x, w0, out, 0);
  // fb1: w=32 s=8  d=128 P=167  -> cols [21632, 43008)
  fb_wmma_kernel<32, 8, 128, 167, 5><<<dim3(167, 8),  32, 0, stream>>>(x, w1, out, 21632);
  // fb2: w=64 s=16 d=256 P=82   -> cols [43008, 64000)
  fb_wmma_kernel<64, 16, 256, 82, 6><<<dim3(82, 16), 32, 0, stream>>>(x, w2, out, 43008);
}